// Net_68375879352646
// MI455X (gfx1250) — compile-verified
//
#include <hip/hip_runtime.h>
#include <math.h>

#define NN   100000
#define EE   3200000
#define K1   602
#define HID  128
#define OC   41
#define OCP  48

typedef __attribute__((ext_vector_type(2))) float v2f;
typedef __attribute__((ext_vector_type(8))) float v8f;

// ---------------- CSR build ----------------

__global__ void k_zero_i32(int* __restrict__ p, int n) {
  int i = blockIdx.x * blockDim.x + threadIdx.x;
  if (i < n) p[i] = 0;
}

__global__ void k_degree(const int* __restrict__ dstv, int* __restrict__ deg) {
  int e = blockIdx.x * blockDim.x + threadIdx.x;
  if (e < EE) atomicAdd(&deg[dstv[e]], 1);
}

__global__ void k_scan(const int* __restrict__ deg, int* __restrict__ row_ptr,
                       int* __restrict__ cursor, int n) {
  __shared__ int sm[1024];
  __shared__ int running;
  int tid = threadIdx.x;
  if (tid == 0) running = 0;
  __syncthreads();
  for (int base = 0; base < n; base += 1024) {
    int i = base + tid;
    int v = (i < n) ? deg[i] : 0;
    sm[tid] = v;
    __syncthreads();
    for (int off = 1; off < 1024; off <<= 1) {
      int t = (tid >= off) ? sm[tid - off] : 0;
      __syncthreads();
      sm[tid] += t;
      __syncthreads();
    }
    int incl = sm[tid];
    int r = running;
    if (i < n) { int e = r + incl - v; row_ptr[i] = e; cursor[i] = e; }
    __syncthreads();
    if (tid == 1023) running = r + sm[1023];
    __syncthreads();
  }
  if (tid == 0) row_ptr[n] = running;
}

__global__ void k_bucket(const int* __restrict__ srcv, const int* __restrict__ dstv,
                         int* __restrict__ cursor, int* __restrict__ esrc) {
  int e = blockIdx.x * blockDim.x + threadIdx.x;
  if (e < EE) {
    int p = atomicAdd(&cursor[dstv[e]], 1);
    esrc[p] = srcv[e];
  }
}

// ---------------- GEMM1: h1l = x @ W1_l^T, h1r = x @ W1_r^T  (fp32 WMMA) ----------------
// Block = 256 thr = 8 waves; each wave owns 16 rows, all 128 output cols, both matrices.

__global__ __launch_bounds__(256) void k_gemm1(
    const float* __restrict__ x, const float* __restrict__ Wl,
    const float* __restrict__ Wr, float* __restrict__ h1l, float* __restrict__ h1r) {
  __shared__ float xs[128 * 34];        // 128 rows x 32 K, pad 34
  __shared__ float wt[2][32 * 130];     // [mat][k][n], 32 K x 128 N, pad 130
  int tid = threadIdx.x;
  int m0 = blockIdx.x * 128;
  int lane = tid & 31, wv = tid >> 5;
  int mloc = lane & 15;
  int kh = (lane >> 4) * 2;             // f32 A/B frag: lane half selects K pair

  v8f acc[2][8];
  v8f z = {0.f, 0.f, 0.f, 0.f, 0.f, 0.f, 0.f, 0.f};
#pragma unroll
  for (int a = 0; a < 2; ++a)
#pragma unroll
    for (int b = 0; b < 8; ++b) acc[a][b] = z;

  for (int kc = 0; kc < K1; kc += 32) {
    for (int idx = tid; idx < 128 * 32; idx += 256) {
      int r = idx >> 5, k = idx & 31;
      int row = m0 + r, kg = kc + k;
      float v = 0.f;
      if (row < NN && kg < K1) v = x[row * K1 + kg];
      xs[r * 34 + k] = v;
    }
    for (int idx = tid; idx < 2 * 128 * 32; idx += 256) {
      int mat = idx >> 12;
      int rem = idx & 4095;
      int nn = rem >> 5, k = rem & 31;
      int kg = kc + k;
      const float* W = mat ? Wr : Wl;
      float v = (kg < K1) ? W[nn * K1 + kg] : 0.f;
      wt[mat][k * 130 + nn] = v;
    }
    __syncthreads();
    for (int kk = 0; kk < 32; kk += 4) {
      v2f a;
      int abase = (wv * 16 + mloc) * 34 + kk + kh;
      a.x = xs[abase];
      a.y = xs[abase + 1];
#pragma unroll
      for (int mat = 0; mat < 2; ++mat) {
#pragma unroll
        for (int nt = 0; nt < 8; ++nt) {
          v2f b;
          int bbase = (kk + kh) * 130 + nt * 16 + mloc;
          b.x = wt[mat][bbase];
          b.y = wt[mat][bbase + 130];
          acc[mat][nt] = __builtin_amdgcn_wmma_f32_16x16x4_f32(
              false, a, false, b, (short)0, acc[mat][nt], false, false);
        }
      }
    }
    __syncthreads();
  }
#pragma unroll
  for (int mat = 0; mat < 2; ++mat) {
    float* o = mat ? h1r : h1l;
#pragma unroll
    for (int nt = 0; nt < 8; ++nt) {
#pragma unroll
      for (int r = 0; r < 8; ++r) {
        int Mrow = m0 + wv * 16 + r + 8 * (lane >> 4);
        int col = nt * 16 + mloc;
        if (Mrow < NN) o[Mrow * HID + col] = acc[mat][nt][r];
      }
    }
  }
}

// ---------------- Layer-1 aggregation + ReLU (one wave per node, no atomics) ----------------

__global__ __launch_bounds__(256) void k_agg1(
    const float* __restrict__ h1l, const float* __restrict__ h1r,
    const float* __restrict__ b1, const int* __restrict__ row_ptr,
    const int* __restrict__ esrc, const int* __restrict__ deg,
    float* __restrict__ hbuf) {
  int wid = (blockIdx.x * blockDim.x + threadIdx.x) >> 5;
  int lane = threadIdx.x & 31;
  if (wid >= NN) return;
  int s0 = row_ptr[wid], s1 = row_ptr[wid + 1];
  float4 acc = make_float4(0.f, 0.f, 0.f, 0.f);
  const float4* base = (const float4*)h1l;   // 32 lanes x float4 = 128 ch row
  for (int j = s0; j < s1; ++j) {
    int s = esrc[j];
    float4 v = base[(size_t)s * 32 + lane];
    acc.x += v.x; acc.y += v.y; acc.z += v.z; acc.w += v.w;
  }
  float invd = 1.f / fmaxf((float)deg[wid], 1.f);
  float4 hr = ((const float4*)h1r)[(size_t)wid * 32 + lane];
  float4 bb = ((const float4*)b1)[lane];
  float4 o;
  o.x = fmaxf(acc.x * invd + bb.x + hr.x, 0.f);
  o.y = fmaxf(acc.y * invd + bb.y + hr.y, 0.f);
  o.z = fmaxf(acc.z * invd + bb.z + hr.z, 0.f);
  o.w = fmaxf(acc.w * invd + bb.w + hr.w, 0.f);
  ((float4*)hbuf)[(size_t)wid * 32 + lane] = o;
}

// ---------------- GEMM2: h2l = h @ W2_l^T, h2r = h @ W2_r^T  (fp32 WMMA, 41->48 pad) -------

__global__ __launch_bounds__(256) void k_gemm2(
    const float* __restrict__ h, const float* __restrict__ Wl,
    const float* __restrict__ Wr, float* __restrict__ h2l, float* __restrict__ h2r) {
  __shared__ float hs[128 * 66];        // 128 rows x 64 K, pad 66
  __shared__ float wt[2][64 * 50];      // [mat][k][n], 64 K x 48 N, pad 50
  int tid = threadIdx.x;
  int m0 = blockIdx.x * 128;
  int lane = tid & 31, wv = tid >> 5;
  int mloc = lane & 15;
  int kh = (lane >> 4) * 2;

  v8f acc[2][3];
  v8f z = {0.f, 0.f, 0.f, 0.f, 0.f, 0.f, 0.f, 0.f};
#pragma unroll
  for (int a = 0; a < 2; ++a)
#pragma unroll
    for (int b = 0; b < 3; ++b) acc[a][b] = z;

  for (int kc = 0; kc < HID; kc += 64) {
    for (int idx = tid; idx < 128 * 64; idx += 256) {
      int r = idx >> 6, k = idx & 63;
      int row = m0 + r;
      float v = 0.f;
      if (row < NN) v = h[(size_t)row * HID + kc + k];
      hs[r * 66 + k] = v;
    }
    for (int idx = tid; idx < 2 * 48 * 64; idx += 256) {
      int mat = idx / 3072;
      int rem = idx - mat * 3072;
      int nn = rem >> 6, k = rem & 63;
      const float* W = mat ? Wr : Wl;
      float v = (nn < OC) ? W[nn * HID + kc + k] : 0.f;
      wt[mat][k * 50 + nn] = v;
    }
    __syncthreads();
    for (int kk = 0; kk < 64; kk += 4) {
      v2f a;
      int abase = (wv * 16 + mloc) * 66 + kk + kh;
      a.x = hs[abase];
      a.y = hs[abase + 1];
#pragma unroll
      for (int mat = 0; mat < 2; ++mat) {
#pragma unroll
        for (int nt = 0; nt < 3; ++nt) {
          v2f b;
          int bbase = (kk + kh) * 50 + nt * 16 + mloc;
          b.x = wt[mat][bbase];
          b.y = wt[mat][bbase + 50];
          acc[mat][nt] = __builtin_amdgcn_wmma_f32_16x16x4_f32(
              false, a, false, b, (short)0, acc[mat][nt], false, false);
        }
      }
    }
    __syncthreads();
  }
#pragma unroll
  for (int mat = 0; mat < 2; ++mat) {
    float* o = mat ? h2r : h2l;
#pragma unroll
    for (int nt = 0; nt < 3; ++nt) {
#pragma unroll
      for (int r = 0; r < 8; ++r) {
        int Mrow = m0 + wv * 16 + r + 8 * (lane >> 4);
        int col = nt * 16 + mloc;
        if (Mrow < NN) o[(size_t)Mrow * OCP + col] = acc[mat][nt][r];
      }
    }
  }
}

// ---------------- Layer-2 aggregation + bias + log_softmax (fused) ----------------

__global__ __launch_bounds__(256) void k_agg2(
    const float* __restrict__ h2l, const float* __restrict__ h2r,
    const float* __restrict__ b2, const int* __restrict__ row_ptr,
    const int* __restrict__ esrc, const int* __restrict__ deg,
    float* __restrict__ out) {
  int wid = (blockIdx.x * blockDim.x + threadIdx.x) >> 5;
  int lane = threadIdx.x & 31;
  if (wid >= NN) return;
  int s0 = row_ptr[wid], s1 = row_ptr[wid + 1];
  float a0 = 0.f, a1 = 0.f;
  for (int j = s0; j < s1; ++j) {
    int s = esrc[j];
    a0 += h2l[(size_t)s * OCP + lane];
    if (lane < 16) a1 += h2l[(size_t)s * OCP + 32 + lane];
  }
  float invd = 1.f / fmaxf((float)deg[wid], 1.f);
  int c0 = lane, c1 = lane + 32;
  float v0 = a0 * invd + ((c0 < OC) ? b2[c0] : 0.f) + h2r[(size_t)wid * OCP + c0];
  float v1 = -3.4e38f;
  if (lane < 16) {
    float bb = (c1 < OC) ? b2[c1] : 0.f;
    v1 = a1 * invd + bb + h2r[(size_t)wid * OCP + c1];
  }
  float m = (c0 < OC) ? v0 : -3.4e38f;
  if (c1 < OC) m = fmaxf(m, v1);
#pragma unroll
  for (int off = 16; off > 0; off >>= 1) m = fmaxf(m, __shfl_xor(m, off, 32));
  float se = ((c0 < OC) ? __expf(v0 - m) : 0.f) + ((c1 < OC) ? __expf(v1 - m) : 0.f);
#pragma unroll
  for (int off = 16; off > 0; off >>= 1) se += __shfl_xor(se, off, 32);
  float lse = __logf(se);
  if (c0 < OC) out[(size_t)wid * OC + c0] = v0 - m - lse;
  if (c1 < OC) out[(size_t)wid * OC + c1] = v1 - m - lse;
}

// ---------------- host ----------------

extern "C" void kernel_launch(void* const* d_in, const int* in_sizes, int n_in,
                              void* d_out, int out_size, void* d_ws, size_t ws_size,
                              hipStream_t stream) {
  (void)in_sizes; (void)n_in; (void)out_size; (void)ws_size;
  const float* x   = (const float*)d_in[0];
  const int*   ei  = (const int*)d_in[1];
  const float* W1l = (const float*)d_in[2];
  const float* b1  = (const float*)d_in[3];
  const float* W1r = (const float*)d_in[4];
  const float* W2l = (const float*)d_in[5];
  const float* b2  = (const float*)d_in[6];
  const float* W2r = (const float*)d_in[7];
  float* out = (float*)d_out;
  const int* srcv = ei;        // edge_index[0]
  const int* dstv = ei + EE;   // edge_index[1]

  char* p = (char*)d_ws;
  auto alloc = [&](size_t bytes) {
    void* r = (void*)p;
    p += (bytes + 255) & ~(size_t)255;
    return r;
  };
  int*   deg     = (int*)alloc((size_t)NN * 4);
  int*   row_ptr = (int*)alloc((size_t)(NN + 1) * 4);
  int*   cursor  = (int*)alloc((size_t)NN * 4);
  int*   esrc    = (int*)alloc((size_t)EE * 4);
  float* h1l     = (float*)alloc((size_t)NN * HID * 4);
  float* h1r     = (float*)alloc((size_t)NN * HID * 4);
  float* hbuf    = (float*)alloc((size_t)NN * HID * 4);
  float* h2l     = (float*)alloc((size_t)NN * OCP * 4);
  float* h2r     = (float*)alloc((size_t)NN * OCP * 4);

  k_zero_i32<<<(NN + 255) / 256, 256, 0, stream>>>(deg, NN);
  k_degree<<<(EE + 255) / 256, 256, 0, stream>>>(dstv, deg);
  k_scan<<<1, 1024, 0, stream>>>(deg, row_ptr, cursor, NN);
  k_bucket<<<(EE + 255) / 256, 256, 0, stream>>>(srcv, dstv, cursor, esrc);
  k_gemm1<<<(NN + 127) / 128, 256, 0, stream>>>(x, W1l, W1r, h1l, h1r);
  k_agg1<<<(NN + 7) / 8, 256, 0, stream>>>(h1l, h1r, b1, row_ptr, esrc, deg, hbuf);
  k_gemm2<<<(NN + 127) / 128, 256, 0, stream>>>(hbuf, W2l, W2r, h2l, h2r);
  k_agg2<<<(NN + 7) / 8, 256, 0, stream>>>(h2l, h2r, b2, row_ptr, esrc, deg, out);
}